// GCN_model_27719718928864
// MI455X (gfx1250) — compile-verified
//
#include <hip/hip_runtime.h>

#define N_NODES 100000
#define N_EDGES 1600000
#define IN_CH 128
#define HID_CH 128
#define OUT_CH 64

typedef float v2f  __attribute__((ext_vector_type(2)));
typedef float v8f  __attribute__((ext_vector_type(8)));
typedef float f4v  __attribute__((ext_vector_type(4)));

// ---------------- utility kernels ----------------
__global__ __launch_bounds__(256) void k_fill(float* __restrict__ p, float v, int n) {
    int i = blockIdx.x * blockDim.x + threadIdx.x;
    if (i < n) p[i] = v;
}

__global__ __launch_bounds__(256) void k_deg_accum(const int* __restrict__ col,
                                                   const float* __restrict__ w,
                                                   float* __restrict__ deg, int E) {
    int e = blockIdx.x * blockDim.x + threadIdx.x;
    if (e < E) atomicAdd(&deg[col[e]], w[e]);
}

__global__ __launch_bounds__(256) void k_rsqrt(float* __restrict__ d, int n) {
    int i = blockIdx.x * blockDim.x + threadIdx.x;
    if (i < n) d[i] = rsqrtf(d[i]);   // deg >= 1 always (self loop), matches reference
}

__global__ __launch_bounds__(256) void k_norm(const int* __restrict__ row,
                                              const int* __restrict__ col,
                                              const float* __restrict__ w,
                                              const float* __restrict__ dinv,
                                              float* __restrict__ norm, int E) {
    int e = blockIdx.x * blockDim.x + threadIdx.x;
    if (e < E) norm[e] = dinv[row[e]] * w[e] * dinv[col[e]];
}

// ---------------- fp32 WMMA GEMM: out[M,NCOL] = A[M,128] * W[128,NCOL] ----------------
// grid.x = M/16 row-tiles; block = (NCOL/16) waves of 32; K = 128 fixed.
template <int NCOL>
__global__ __launch_bounds__((NCOL / 16) * 32)
void k_gemm_wmma(const float* __restrict__ A, const float* __restrict__ W,
                 float* __restrict__ out) {
    const int mtile = blockIdx.x;
    const int wv    = threadIdx.x >> 5;   // N-tile owned by this wave
    const int lane  = threadIdx.x & 31;
    const int half  = lane >> 4;          // 0: K pair {k,k+1}, 1: {k+2,k+3}
    const int lm    = lane & 15;

    const float* __restrict__ arow = A + (size_t)(mtile * 16 + lm) * 128;
    const int bcol = wv * 16 + lm;

    v8f c = {};
#pragma unroll 8
    for (int k0 = 0; k0 < 128; k0 += 4) {
        const int ka = k0 + 2 * half;
        v2f av; av.x = arow[ka];              av.y = arow[ka + 1];
        v2f bv; bv.x = W[ka * NCOL + bcol];   bv.y = W[(ka + 1) * NCOL + bcol];
        // D = A*B + C, full fp32 precision: v_wmma_f32_16x16x4_f32
        c = __builtin_amdgcn_wmma_f32_16x16x4_f32(false, av, false, bv,
                                                  (short)0, c, false, false);
    }
    // C/D layout: VGPR r -> row (r + 8*half), lane lm -> col
    float* __restrict__ op = out + (size_t)(mtile * 16 + 8 * half) * NCOL + bcol;
#pragma unroll
    for (int r = 0; r < 8; ++r) op[r * NCOL] = c[r];
}

// ---------------- edge aggregation: agg[col] += norm * h[row] ----------------
// One (edge, 4-channel group) per thread; C/4 lanes cooperate on an edge, so
// row/col/norm loads are wave-uniform (broadcast) and h gathers are float4.
template <int C>
__global__ __launch_bounds__(256)
void k_aggregate(const float* __restrict__ h, const int* __restrict__ row,
                 const int* __restrict__ col, const float* __restrict__ norm,
                 float* __restrict__ agg, int E) {
    const int lpe = C / 4;
    long long t = (long long)blockIdx.x * blockDim.x + threadIdx.x;
    int e = (int)(t / lpe);
    int g = (int)(t % lpe);
    if (e >= E) return;
    const float nn = norm[e];
    const int r = row[e];
    const int d = col[e];
    f4v v = *((const f4v*)(h + (size_t)r * C) + g);
    float* __restrict__ ap = agg + (size_t)d * C + g * 4;
    atomicAdd(ap + 0, nn * v.x);
    atomicAdd(ap + 1, nn * v.y);
    atomicAdd(ap + 2, nn * v.z);
    atomicAdd(ap + 3, nn * v.w);
}

// ---------------- self-loop + bias (+ optional ReLU), in place ----------------
template <int C>
__global__ __launch_bounds__(256)
void k_finalize(float* __restrict__ agg, const float* __restrict__ h,
                const float* __restrict__ dinv, const float* __restrict__ bias,
                int relu) {
    long long i = (long long)blockIdx.x * blockDim.x + threadIdx.x;
    if (i >= (long long)N_NODES * C) return;
    int node = (int)(i / C);
    int ch   = (int)(i % C);
    float di = dinv[node];
    float v = agg[i] + di * di * h[i] + bias[ch];
    if (relu) v = fmaxf(v, 0.0f);
    agg[i] = v;
}

extern "C" void kernel_launch(void* const* d_in, const int* in_sizes, int n_in,
                              void* d_out, int out_size, void* d_ws, size_t ws_size,
                              hipStream_t stream) {
    const float* x  = (const float*)d_in[0];
    const int*   ei = (const int*)d_in[1];
    const float* ew = (const float*)d_in[2];
    const float* W1 = (const float*)d_in[3];
    const float* b1 = (const float*)d_in[4];
    const float* W2 = (const float*)d_in[5];
    const float* b2 = (const float*)d_in[6];
    const int* row = ei;             // edge_index[0] : sources
    const int* col = ei + N_EDGES;   // edge_index[1] : destinations

    char* ws = (char*)d_ws;
    float* dinv = (float*)(ws);                    //  0.4 MB (deg, then rsqrt in place)
    float* norm = (float*)(ws + 524288ULL);        //  6.4 MB
    float* h1   = (float*)(ws + 8912896ULL);       // 51.2 MB
    float* agg1 = (float*)(ws + 61341696ULL);      // 51.2 MB
    float* h2   = (float*)(ws + 113770496ULL);     // 25.6 MB
    float* out  = (float*)d_out;

    // ---- gcn_norm ----
    k_fill<<<(N_NODES + 255) / 256, 256, 0, stream>>>(dinv, 1.0f, N_NODES); // self loops
    k_deg_accum<<<(N_EDGES + 255) / 256, 256, 0, stream>>>(col, ew, dinv, N_EDGES);
    k_rsqrt<<<(N_NODES + 255) / 256, 256, 0, stream>>>(dinv, N_NODES);
    k_norm<<<(N_EDGES + 255) / 256, 256, 0, stream>>>(row, col, ew, dinv, norm, N_EDGES);

    // ---- layer 1: h1 = x @ W1 (WMMA fp32); agg1 = scatter(norm*h1) ; relu+bias+selfloop ----
    k_gemm_wmma<HID_CH><<<N_NODES / 16, (HID_CH / 16) * 32, 0, stream>>>(x, W1, h1);
    k_fill<<<(N_NODES * HID_CH + 255) / 256, 256, 0, stream>>>(agg1, 0.0f, N_NODES * HID_CH);
    k_aggregate<HID_CH><<<(unsigned)(((long long)N_EDGES * (HID_CH / 4) + 255) / 256), 256, 0, stream>>>(
        h1, row, col, norm, agg1, N_EDGES);
    k_finalize<HID_CH><<<(unsigned)(((long long)N_NODES * HID_CH + 255) / 256), 256, 0, stream>>>(
        agg1, h1, dinv, b1, 1);

    // ---- layer 2: h2 = agg1 @ W2 (WMMA fp32); aggregate into d_out; finalize in place ----
    k_gemm_wmma<OUT_CH><<<N_NODES / 16, (OUT_CH / 16) * 32, 0, stream>>>(agg1, W2, h2);
    k_fill<<<(N_NODES * OUT_CH + 255) / 256, 256, 0, stream>>>(out, 0.0f, N_NODES * OUT_CH);
    k_aggregate<OUT_CH><<<(unsigned)(((long long)N_EDGES * (OUT_CH / 4) + 255) / 256), 256, 0, stream>>>(
        h2, row, col, norm, out, N_EDGES);
    k_finalize<OUT_CH><<<(unsigned)(((long long)N_NODES * OUT_CH + 255) / 256), 256, 0, stream>>>(
        out, h2, dinv, b2, 0);
}